// HandmadeConv2D_1116691497000
// MI455X (gfx1250) — compile-verified
//
#include <hip/hip_runtime.h>

// CDNA5 / gfx1250 implicit-GEMM conv: V_WMMA_F32_16X16X4_F32 (wave32).
// K permuted to k' = tap*32 + ci; x tile stored ci-pair-interleaved so each
// lane's B k-pair is one 8B-aligned float2 -> ds_load_b64 with imm offset.
// Async global->LDS staging (ASYNCcnt path).
typedef float v2f __attribute__((ext_vector_type(2)));
typedef float v8f __attribute__((ext_vector_type(8)));

#define B_      16
#define CI_     32
#define H_      224
#define W_      224
#define CO_     32
#define KS_     3
#define OH_     222
#define OW_     222
#define KDIM_   (CI_ * KS_ * KS_)     // 288
#define WSTR_   292                   // padded A row stride (bank-spread: 292%64=36)
#define NT_     32                    // output columns per block
#define OHT_    2                     // output rows per block
#define XROWS_  (OHT_ + KS_ - 1)      // 4 staged input rows
#define XCP_    36                    // padded staged row length (needs 34)
#define XELEMS_ (CI_ * XROWS_ * XCP_) // 4608 floats = 18 KB
#define PSTB_   (XROWS_ * XCP_ * 8)   // 1152 B: byte stride between ci-PAIR planes
#define THREADS_ 256                  // 8 wave32 waves

#if defined(__has_builtin)
#if __has_builtin(__builtin_amdgcn_global_load_async_to_lds_b32)
#define HAVE_ASYNC_LDS 1
#endif
#endif

typedef __attribute__((address_space(1))) int  g_int;
typedef __attribute__((address_space(3))) int  l_int;

__launch_bounds__(THREADS_, 2)
__global__ void conv2d_wmma_f32_kernel(const float* __restrict__ x,
                                       const float* __restrict__ w,
                                       const float* __restrict__ bias,
                                       float* __restrict__ out)
{
    // 36.5 KB permuted weights + 18 KB interleaved x tile + bias (~55 KB).
    __shared__ __align__(16) float lds_wA[CO_ * WSTR_]; // A[co][k'=tap*32+ci]
    __shared__ __align__(16) float xsd[XELEMS_];        // x[ci/2][row][col][2]
    __shared__ __align__(16) float lds_bias[CO_];

    const int tid = threadIdx.x;
    const int owt = blockIdx.x;            // 0..6
    const int oh0 = blockIdx.y * OHT_;     // 0,2,..,220
    const int b   = blockIdx.z;            // 0..15
    const int owbase = owt * NT_;

    // ---- Stage permuted weights and ci-interleaved x tile into LDS ----
#if HAVE_ASYNC_LDS
    for (int i = tid; i < CO_ * KDIM_; i += THREADS_) {
        const int co = i / KDIM_, r = i % KDIM_;
        const int tap = r >> 5, ci = r & 31;          // k' decode
        __builtin_amdgcn_global_load_async_to_lds_b32(
            (g_int*)(w + co * KDIM_ + ci * (KS_ * KS_) + tap),
            (l_int*)(&lds_wA[co * WSTR_ + r]), 0, 0);
    }
    for (int i = tid; i < XELEMS_; i += THREADS_) {
        // dst linear index decode for layout [p][R][C][2]:
        const int j  = i & 1;
        const int t  = i >> 1;
        const int C  = t % XCP_;
        const int t2 = t / XCP_;
        const int R  = t2 & (XROWS_ - 1);
        const int p  = t2 >> 2;
        const int ci = 2 * p + j;
        const int gc  = owbase + C;                    // clamp to W-1
        const int gcc = gc < (W_ - 1) ? gc : (W_ - 1);
        __builtin_amdgcn_global_load_async_to_lds_b32(
            (g_int*)(x + (((size_t)b * CI_ + ci) * H_ + (oh0 + R)) * W_ + gcc),
            (l_int*)(&xsd[i]), 0, 0);
    }
#else
    for (int i = tid; i < CO_ * KDIM_; i += THREADS_) {
        const int co = i / KDIM_, r = i % KDIM_;
        const int tap = r >> 5, ci = r & 31;
        lds_wA[co * WSTR_ + r] = w[co * KDIM_ + ci * (KS_ * KS_) + tap];
    }
    for (int i = tid; i < XELEMS_; i += THREADS_) {
        const int j  = i & 1;
        const int t  = i >> 1;
        const int C  = t % XCP_;
        const int t2 = t / XCP_;
        const int R  = t2 & (XROWS_ - 1);
        const int p  = t2 >> 2;
        const int ci = 2 * p + j;
        const int gc  = owbase + C;
        const int gcc = gc < (W_ - 1) ? gc : (W_ - 1);
        xsd[i] = x[(((size_t)b * CI_ + ci) * H_ + (oh0 + R)) * W_ + gcc];
    }
#endif
    if (tid < CO_) lds_bias[tid] = bias[tid];

#if HAVE_ASYNC_LDS
#if __has_builtin(__builtin_amdgcn_s_wait_asynccnt)
    __builtin_amdgcn_s_wait_asynccnt(0);
#else
    asm volatile("s_wait_asynccnt 0x0" ::: "memory");
#endif
#endif
    __syncthreads();

    // ---- Wave tiling: 2 M-tiles x 2 N-subtiles x 2 output rows ----
    const int lane  = tid & 31;
    const int wave  = tid >> 5;
    const int mtile = wave & 1;          // CO rows [0,16) / [16,32)
    const int ntile = (wave >> 1) & 1;   // 16-column subtile
    const int ohsub = wave >> 2;         // output row within block
    const int col   = lane & 15;
    const int half  = lane >> 4;         // selects K-pair (A/B 32-bit layout)

    // A: lane row = mtile*16+col; half folds the k-pair select (+2 floats).
    const float* __restrict__ aw =
        &lds_wA[(mtile * 16 + col) * WSTR_] + half * 2;
    // B: per-lane base; half selects the next ci-pair plane (+1152 B).
    const char* __restrict__ bx =
        (const char*)xsd + (ohsub * XCP_ + ntile * 16 + col) * 8 + half * PSTB_;

    // Byte offset of tap (kh,kw) inside a ci-pair plane: kh*288 + kw*8.
    constexpr int TB2[9] = {
        0 * (XCP_ * 8) + 0 * 8, 0 * (XCP_ * 8) + 1 * 8, 0 * (XCP_ * 8) + 2 * 8,
        1 * (XCP_ * 8) + 0 * 8, 1 * (XCP_ * 8) + 1 * 8, 1 * (XCP_ * 8) + 2 * 8,
        2 * (XCP_ * 8) + 0 * 8, 2 * (XCP_ * 8) + 1 * 8, 2 * (XCP_ * 8) + 2 * 8 };

    v8f acc = {0.f, 0.f, 0.f, 0.f, 0.f, 0.f, 0.f, 0.f};

    // 72 fully-unrolled k'-steps: one b64 A + one b64 B + one WMMA, all imm.
    #pragma unroll
    for (int k0 = 0; k0 < KDIM_; k0 += 4) {
        const int tap  = k0 >> 5;                       // constant after unroll
        const int boff = TB2[tap] + ((k0 & 31) >> 1) * PSTB_;
        v2f a = *(const v2f*)(aw + k0);                 // ds_load_b64 (2addr-merged)
        v2f bfrag = *(const v2f*)(bx + boff);           // ds_load_b64, imm offset
        acc = __builtin_amdgcn_wmma_f32_16x16x4_f32(
            false, a, false, bfrag, (short)0, acc, false, false);
    }

    // ---- Epilogue: C/D layout VGPR j -> M=j (lanes 0-15), M=j+8 (16-31) ----
    const int oh = oh0 + ohsub;
    const int ow = owbase + ntile * 16 + col;
    if (ow < OW_) {
        #pragma unroll
        for (int j = 0; j < 8; ++j) {
            const int co = mtile * 16 + half * 8 + j;
            const size_t oidx = (((size_t)b * CO_ + co) * OH_ + oh) * (size_t)OW_ + ow;
            out[oidx] = acc[j] + lds_bias[co];
        }
    }
}

extern "C" void kernel_launch(void* const* d_in, const int* in_sizes, int n_in,
                              void* d_out, int out_size, void* d_ws, size_t ws_size,
                              hipStream_t stream) {
    const float* x    = (const float*)d_in[0];
    const float* w    = (const float*)d_in[1];
    const float* bias = (const float*)d_in[2];
    float* out        = (float*)d_out;

    dim3 grid((OW_ + NT_ - 1) / NT_, OH_ / OHT_, B_);  // (7, 111, 16)
    conv2d_wmma_f32_kernel<<<grid, THREADS_, 0, stream>>>(x, w, bias, out);
}